// H_RevGNN_56126632624668
// MI455X (gfx1250) — compile-verified
//
#include <hip/hip_runtime.h>
#include <hip/hip_bf16.h>

// ---------------------------------------------------------------------------
// RevGNN forward for MI455X (gfx1250), fp32 end-to-end.
//   h = x @ lin1_w + lin1_b
//   8x { z = relu(LN(y_in)) @ conv_w ; h_g += conv_b ; h_g += scatter(w * z[src]) }
//   out = relu(LN(h)) @ lin2_w + lin2_b
// GEMMs use V_WMMA_F32_16X16X4_F32 (full fp32 matrix pipe); LayerNorm+ReLU is
// fused into the WMMA A-fragment path using precomputed per-row (mu, rstd).
// The k-loop is branch-free (lin2 weight zero-padded to 64 cols in workspace)
// and explicitly double-buffered: stage-(k+4) fragment loads are issued while
// stage-k WMMAs execute, so no WMMA waits on a just-issued load.
// Edge aggregation: one wave per edge, float4 gather from L2-resident z,
// float atomicAdd scatter into h (reversible update done in place).
// ---------------------------------------------------------------------------

typedef __attribute__((ext_vector_type(2))) float v2f;
typedef __attribute__((ext_vector_type(8))) float v8f;

#define WAVES_PER_BLOCK 8   // 8 waves x 16 rows = 128 rows per block
#define NTILE 4             // 4 x 16 = 64 output columns per wave

__device__ __forceinline__ v8f wmma_f32_16x16x4(v2f a, v2f b, v8f c) {
    // 8 args: (neg_a, A, neg_b, B, c_mod, C, reuse_a, reuse_b)
    return __builtin_amdgcn_wmma_f32_16x16x4_f32(
        false, a, false, b, (short)0, c, false, false);
}

// Per-row LayerNorm statistics: stats[2*row] = mean, stats[2*row+1] = rsqrt(var+eps).
// One wave per row; cols is 128 or 256; lda is the row stride of A.
__global__ __launch_bounds__(256) void row_stats_kernel(
    const float* __restrict__ A, int lda, int cols,
    float* __restrict__ stats, int nrows)
{
    const int lane = threadIdx.x & 31;
    const int row  = blockIdx.x * 8 + (threadIdx.x >> 5);
    if (row >= nrows) return;
    const float* a = A + (size_t)row * lda;
    float s = 0.f, s2 = 0.f;
    for (int c = lane * 4; c < cols; c += 128) {
        const float4 v = *(const float4*)(a + c);
        s  += v.x + v.y + v.z + v.w;
        s2 += v.x * v.x + v.y * v.y + v.z * v.z + v.w * v.w;
    }
    #pragma unroll
    for (int off = 16; off > 0; off >>= 1) {
        s  += __shfl_xor(s,  off, 32);
        s2 += __shfl_xor(s2, off, 32);
    }
    const float inv = 1.f / (float)cols;
    const float mu  = s * inv;
    const float var = s2 * inv - mu * mu;
    if (lane == 0) {
        stats[2 * row]     = mu;
        stats[2 * row + 1] = rsqrtf(var + 1e-5f);
    }
}

// One k-step's worth of operand fragments (K advances by 4 per stage).
struct KFrag {
    v2f a;          // A[m][ka..ka+1]
    v2f b[NTILE];   // B[ka..ka+1][n0 + t*16 + mlane]
    v2f g, be;      // gamma/beta[ka..ka+1] (LN path only)
};

// C[M, ncols_out] = act(A[M,K]) @ B[K, >= gridDim.y*64] (+ bias),
// act = relu(LN(.)) if FUSE_LN.  B must be loadable (padded) out to the full
// 64-column wave tile: the k-loop is branch-free by construction.  The real
// output width ncols_out only guards the epilogue stores.
// Wave tile: 16 rows x 64 cols; K stepped by 4 via V_WMMA_F32_16X16X4_F32,
// software-pipelined two stages deep.
template <int KDIM, bool FUSE_LN>
__global__ __launch_bounds__(256) void gemm_ln_kernel(
    const float* __restrict__ A, int lda,
    const float* __restrict__ stats,
    const float* __restrict__ gamma, const float* __restrict__ beta,
    const float* __restrict__ B, int ldb, int ncols_out,
    const float* __restrict__ bias,
    float* __restrict__ C, int ldc, int nrows)
{
    const int lane  = threadIdx.x & 31;
    const int wave  = threadIdx.x >> 5;
    const int m0    = (blockIdx.x * WAVES_PER_BLOCK + wave) * 16;
    if (m0 >= nrows) return;                 // wave-uniform: EXEC stays all-1s
    const int n0    = blockIdx.y * (NTILE * 16);
    const int mlane = lane & 15;
    const int hi    = lane >> 4;             // K-half selector for A/B fragments

    int m = m0 + mlane;
    if (m >= nrows) m = nrows - 1;           // clamp (keeps all lanes live)

    float mu = 0.f, rs = 0.f;
    if (FUSE_LN) { mu = stats[2 * m]; rs = stats[2 * m + 1]; }
    const float* arow = A + (size_t)m * lda;
    const float* bcol = B + n0 + mlane;      // this lane's B column

    v8f acc[NTILE] = {};

    // Issue all loads for one k-stage together (A pair, 4 B pairs, gamma/beta).
    auto load_frag = [&](KFrag& f, int k0) {
        const int ka = k0 + 2 * hi;          // this lane's K pair: ka, ka+1
        f.a.x = arow[ka];
        f.a.y = arow[ka + 1];
        if (FUSE_LN) {
            f.g.x  = gamma[ka];
            f.g.y  = gamma[ka + 1];
            f.be.x = beta[ka];
            f.be.y = beta[ka + 1];
        }
        const float* brow = bcol + (size_t)ka * ldb;
        #pragma unroll
        for (int t = 0; t < NTILE; ++t) {
            f.b[t].x = brow[t * 16];
            f.b[t].y = brow[(size_t)ldb + t * 16];
        }
    };

    // LN math + 4 WMMAs on a staged fragment.
    auto consume = [&](const KFrag& f) {
        v2f a = f.a;
        if (FUSE_LN) {
            a.x = fmaxf((a.x - mu) * rs * f.g.x + f.be.x, 0.f);
            a.y = fmaxf((a.y - mu) * rs * f.g.y + f.be.y, 0.f);
        }
        #pragma unroll
        for (int t = 0; t < NTILE; ++t)
            acc[t] = wmma_f32_16x16x4(a, f.b[t], acc[t]);
    };

    KFrag f0, f1;
    load_frag(f0, 0);
    #pragma unroll 1
    for (int k0 = 0; k0 < KDIM; k0 += 8) {   // KDIM is a multiple of 8
        load_frag(f1, k0 + 4);               // in flight during consume(f0)
        consume(f0);
        if (k0 + 8 < KDIM) load_frag(f0, k0 + 8);  // in flight during consume(f1)
        consume(f1);
    }

    #pragma unroll
    for (int t = 0; t < NTILE; ++t) {
        const int n = n0 + t * 16 + mlane;
        if (n >= ncols_out) continue;
        const float bv = bias ? bias[n] : 0.f;
        #pragma unroll
        for (int r = 0; r < 8; ++r) {        // C layout: row = r + 8*hi
            const int row = m0 + r + 8 * hi;
            if (row < nrows) C[(size_t)row * ldc + n] = acc[t][r] + bv;
        }
    }
}

// Zero-pad lin2_w [256,40] -> [256,64] and lin2_b [40] -> [64] in workspace.
__global__ __launch_bounds__(256) void pad_lin2_kernel(
    const float* __restrict__ w, const float* __restrict__ b,
    float* __restrict__ w_pad, float* __restrict__ b_pad)
{
    const int tid = blockIdx.x * blockDim.x + threadIdx.x;   // 256*64 threads
    if (tid < 64) b_pad[tid] = (tid < 40) ? b[tid] : 0.f;
    const int k = tid >> 6;
    const int n = tid & 63;
    if (k < 256) w_pad[k * 64 + n] = (n < 40) ? w[k * 40 + n] : 0.f;
}

// h_g[row, 0..127] += conv_b   (h_g points at h + g*128, physical stride 256)
__global__ __launch_bounds__(256) void add_bias_kernel(
    float* __restrict__ hg, const float* __restrict__ bias, int nrows)
{
    const int tid = blockIdx.x * blockDim.x + threadIdx.x;
    const int row = tid >> 5;
    if (row >= nrows) return;
    const int c = (tid & 31) * 4;
    const float4 b = *(const float4*)(bias + c);
    float4* p = (float4*)(hg + (size_t)row * 256 + c);
    float4 v = *p;
    v.x += b.x; v.y += b.y; v.z += b.z; v.w += b.w;
    *p = v;
}

// h_g[dst] += w[e] * z[src]; one wave per edge, 4 channels per lane (128 total).
__global__ __launch_bounds__(256) void scatter_kernel(
    const float* __restrict__ z, const int* __restrict__ ei,
    const float* __restrict__ ew, float* __restrict__ hg, int E)
{
    const int tid = blockIdx.x * blockDim.x + threadIdx.x;
    const int e = tid >> 5;
    if (e >= E) return;
    const int c = (tid & 31) * 4;
    const int s = ei[e];
    const int d = ei[E + e];
    const float w = ew[e];
    const float4 v = *(const float4*)(z + (size_t)s * 128 + c);
    float* o = hg + (size_t)d * 256 + c;
    atomicAdd(o + 0, w * v.x);
    atomicAdd(o + 1, w * v.y);
    atomicAdd(o + 2, w * v.z);
    atomicAdd(o + 3, w * v.w);
}

extern "C" void kernel_launch(void* const* d_in, const int* in_sizes, int n_in,
                              void* d_out, int out_size, void* d_ws, size_t ws_size,
                              hipStream_t stream)
{
    (void)n_in; (void)out_size; (void)ws_size;

    const float* x       = (const float*)d_in[0];
    const int*   ei      = (const int*)d_in[1];    // [2, E]
    const float* ew      = (const float*)d_in[2];  // [E]
    const float* lin1_w  = (const float*)d_in[3];  // [128, 256]
    const float* lin1_b  = (const float*)d_in[4];  // [256]
    const float* lin2_w  = (const float*)d_in[5];  // [256, 40]
    const float* lin2_b  = (const float*)d_in[6];  // [40]
    const float* norm_g  = (const float*)d_in[7];  // [4, 2, 128]
    const float* norm_b  = (const float*)d_in[8];
    const float* conv_w  = (const float*)d_in[9];  // [4, 2, 128, 128]
    const float* conv_b  = (const float*)d_in[10]; // [4, 2, 128]
    const float* fnorm_g = (const float*)d_in[11]; // [256]
    const float* fnorm_b = (const float*)d_in[12];
    float* out = (float*)d_out;

    const int N = in_sizes[0] / 128;
    const int E = in_sizes[2];

    // Workspace: h [N,256] | z [N,128] | stats [N,2] | w2_pad [256,64] | b2_pad [64]
    float* h      = (float*)d_ws;
    float* z      = h + (size_t)N * 256;
    float* stats  = z + (size_t)N * 128;
    float* w2_pad = stats + (size_t)N * 2;
    float* b2_pad = w2_pad + 256 * 64;

    const int gemm_bx  = (N + 16 * WAVES_PER_BLOCK - 1) / (16 * WAVES_PER_BLOCK);
    const int stats_bx = (N + 7) / 8;
    const int rowsw_bx = (int)(((long long)N * 32 + 255) / 256);
    const int scat_bx  = (int)(((long long)E * 32 + 255) / 256);

    // 0) pad lin2 weights/bias so the lin2 GEMM inner loop is branch-free
    pad_lin2_kernel<<<(256 * 64) / 256, 256, 0, stream>>>(
        lin2_w, lin2_b, w2_pad, b2_pad);

    // 1) h = x @ lin1_w + lin1_b
    gemm_ln_kernel<128, false><<<dim3(gemm_bx, 4), 256, 0, stream>>>(
        x, 128, nullptr, nullptr, nullptr, lin1_w, 256, 256, lin1_b, h, 256, N);

    // 2) reversible GCN blocks, updated in place in h
    for (int l = 0; l < 4; ++l) {
        for (int g = 0; g < 2; ++g) {
            const int og = 1 - g;          // group the block reads (coupling input)
            const int bi = l * 2 + g;
            row_stats_kernel<<<stats_bx, 256, 0, stream>>>(
                h + og * 128, 256, 128, stats, N);
            gemm_ln_kernel<128, true><<<dim3(gemm_bx, 2), 256, 0, stream>>>(
                h + og * 128, 256, stats, norm_g + bi * 128, norm_b + bi * 128,
                conv_w + (size_t)bi * 128 * 128, 128, 128, nullptr, z, 128, N);
            add_bias_kernel<<<rowsw_bx, 256, 0, stream>>>(
                h + g * 128, conv_b + bi * 128, N);
            scatter_kernel<<<scat_bx, 256, 0, stream>>>(
                z, ei, ew, h + g * 128, E);
        }
    }

    // 3) out = relu(LN(h)) @ lin2_w + lin2_b   (via zero-padded 64-col weight)
    row_stats_kernel<<<stats_bx, 256, 0, stream>>>(h, 256, 256, stats, N);
    gemm_ln_kernel<256, true><<<dim3(gemm_bx, 1), 256, 0, stream>>>(
        h, 256, stats, fnorm_g, fnorm_b, w2_pad, 64, 40, b2_pad, out, 40, N);
}